// LinearLogitLayer_70626442215883
// MI455X (gfx1250) — compile-verified
//
#include <hip/hip_runtime.h>

#define NSPARSE 20
#define NVARLEN 3
#define SEQLEN  50
#define VOCAB   100000
#define NDENSE  13

typedef __attribute__((ext_vector_type(2))) float v2f;
typedef __attribute__((ext_vector_type(8))) float v8f;
typedef __attribute__((ext_vector_type(4))) int   v4i;
typedef __attribute__((ext_vector_type(2))) int   v2i;

// ---------------------------------------------------------------------------
// Kernel 1: dense partial  out[b] = dense_x[b,:] @ dense_w  via
// V_WMMA_F32_16X16X4_F32. One wave handles 16 rows (M). K=13 padded to 16,
// processed as 4 chained K=4 WMMAs. B-matrix holds dense_w broadcast across
// all 16 columns, so every column of D equals the matvec result.
//
// A-layout (ISA 7.12.2, 32-bit A 16x4): lanes 0-15 = M, VGPR0/1 = K0/K1;
// lanes 16-31 = M, VGPR0/1 = K2/K3.  => lane half h owns k = 4c + 2h + {0,1}.
// D-layout (32-bit C/D 16x16): VGPR r, lanes 0-15 -> M=r, lanes 16-31 -> M=8+r.
// ---------------------------------------------------------------------------
__global__ __launch_bounds__(256) void dense_matvec_wmma(
    const float* __restrict__ x, const float* __restrict__ w,
    float* __restrict__ out, int nrows)
{
  const int lane = threadIdx.x & 31;
  const int wave = (blockIdx.x * blockDim.x + threadIdx.x) >> 5;
  const int rowBase = wave * 16;
  if (rowBase >= nrows) return;                    // wave-uniform: EXEC stays all-1s
  const int  m   = lane & 15;
  const int  h   = lane >> 4;                      // half: lanes 16-31 hold K=2,3
  const long row = (long)(rowBase + m);

  v8f acc = {};
#pragma unroll
  for (int c = 0; c < 4; ++c) {
    const int k0 = 4 * c + 2 * h;
    const int k1 = k0 + 1;
    v2f a, bw;
    a.x  = (k0 < NDENSE) ? __builtin_nontemporal_load(&x[row * NDENSE + k0]) : 0.0f;
    a.y  = (k1 < NDENSE) ? __builtin_nontemporal_load(&x[row * NDENSE + k1]) : 0.0f;
    bw.x = (k0 < NDENSE) ? w[k0] : 0.0f;           // broadcast over all 16 columns
    bw.y = (k1 < NDENSE) ? w[k1] : 0.0f;
    acc = __builtin_amdgcn_wmma_f32_16x16x4_f32(false, a, false, bw,
                                                (short)0, acc, false, false);
  }

  // Every column of D is the result; one lane per half (N==0) writes 8 rows.
  if ((lane & 15) == 0) {
#pragma unroll
    for (int r = 0; r < 8; ++r)
      out[rowBase + 8 * h + r] = acc[r];
  }
}

// ---------------------------------------------------------------------------
// Kernel 2: one thread per sample. Streams ids with NT hint (one-shot 44.5MB,
// keep it out of the table-resident L2 working set), random 4B gathers hit
// the 9.2MB L2-resident tables. 4 partial accumulators to relax the FP chain.
// Accumulates onto the dense partial already sitting in out[b].
// ---------------------------------------------------------------------------
__global__ __launch_bounds__(256) void gather_sum_kernel(
    const int*   __restrict__ sparse_ids,
    const int*   __restrict__ varlen_ids,
    const float* __restrict__ sparse_tables,
    const float* __restrict__ varlen_tables,
    float* __restrict__ out, int nb)
{
  const int b = blockIdx.x * blockDim.x + threadIdx.x;
  if (b >= nb) return;

  float s0 = 0.0f, s1 = 0.0f, s2 = 0.0f, s3 = 0.0f;

  // --- 20 sparse gathers; id rows are 80B => 16B aligned => b128 loads ---
  const v4i* sp = (const v4i*)(sparse_ids + (long)b * NSPARSE);
#pragma unroll
  for (int i = 0; i < NSPARSE / 4; ++i) {
    v4i id4 = __builtin_nontemporal_load(sp + i);
    s0 += sparse_tables[(long)(4 * i + 0) * VOCAB + id4.x];
    s1 += sparse_tables[(long)(4 * i + 1) * VOCAB + id4.y];
    s2 += sparse_tables[(long)(4 * i + 2) * VOCAB + id4.z];
    s3 += sparse_tables[(long)(4 * i + 3) * VOCAB + id4.w];
  }

  // --- 3 x 50 masked gathers; rows are 200B => only 8B aligned => b64 loads ---
#pragma unroll
  for (int f = 0; f < NVARLEN; ++f) {
    const float* tf = varlen_tables + (long)f * VOCAB;
    const v2i*   vp = (const v2i*)(varlen_ids + ((long)b * NVARLEN + f) * SEQLEN);
#pragma unroll 5
    for (int i = 0; i < SEQLEN / 2; ++i) {
      v2i id2 = __builtin_nontemporal_load(vp + i);
      if (id2.x != 0) s0 += tf[id2.x];
      if (id2.y != 0) s1 += tf[id2.y];
    }
  }

  out[b] = ((s0 + s1) + (s2 + s3)) + out[b];
}

extern "C" void kernel_launch(void* const* d_in, const int* in_sizes, int n_in,
                              void* d_out, int out_size, void* d_ws, size_t ws_size,
                              hipStream_t stream) {
  (void)in_sizes; (void)n_in; (void)d_ws; (void)ws_size;
  const int*   sparse_ids    = (const int*)d_in[0];
  const int*   varlen_ids    = (const int*)d_in[1];
  const float* dense_x       = (const float*)d_in[2];
  const float* sparse_tables = (const float*)d_in[3];
  const float* varlen_tables = (const float*)d_in[4];
  const float* dense_w       = (const float*)d_in[5];
  float*       out           = (float*)d_out;

  const int nb = out_size;                       // B = 65536

  // Kernel 1: 16 rows per wave, 8 waves per block => 128 rows/block.
  const int blocks1 = (nb + 127) / 128;
  dense_matvec_wmma<<<blocks1, 256, 0, stream>>>(dense_x, dense_w, out, nb);

  // Kernel 2: one thread per sample.
  const int blocks2 = (nb + 255) / 256;
  gather_sum_kernel<<<blocks2, 256, 0, stream>>>(sparse_ids, varlen_ids,
                                                 sparse_tables, varlen_tables,
                                                 out, nb);
}